// PositionalEncoderGrid_25529285607575
// MI455X (gfx1250) — compile-verified
//
#include <hip/hip_runtime.h>
#include <stdint.h>

#ifndef __has_builtin
#define __has_builtin(x) 0
#endif

#define NLEVELS 16
#define TBL     16384          // T = 2^14 entries per level
#define NFEAT   2
#define BLOCK   256            // 8 wave32 waves
#define PPB     16384          // points per block tile

// Typed address-space pointers for the gfx1250 async-LDS DMA builtin.
typedef int v4i __attribute__((ext_vector_type(4)));
typedef __attribute__((address_space(1))) v4i* as1_v4i_ptr;   // global
typedef __attribute__((address_space(3))) v4i* as3_v4i_ptr;   // LDS

// NS[i] = int(16 * 2^(i/3)) as float; W = float(1/NS) (f32-rounded reciprocals)
__constant__ float NS_F[NLEVELS] = {
  16.f, 20.f, 25.f, 32.f, 40.f, 50.f, 64.f, 80.f,
  101.f, 128.f, 161.f, 203.f, 256.f, 322.f, 406.f, 512.f };
__constant__ float W_F[NLEVELS] = {
  1.f/16.f, 1.f/20.f, 1.f/25.f, 1.f/32.f, 1.f/40.f, 1.f/50.f, 1.f/64.f, 1.f/80.f,
  1.f/101.f, 1.f/128.f, 1.f/161.f, 1.f/203.f, 1.f/256.f, 1.f/322.f, 1.f/406.f, 1.f/512.f };

__global__ __launch_bounds__(BLOCK)
void hashgrid_encode_kernel(const float* __restrict__ inputs,
                            const float* __restrict__ table,
                            float* __restrict__ out, int npts)
{
  extern __shared__ float2 lds_tab[];   // 16384 float2 = 128 KB (CDNA5: 320 KB/WGP)

  const int lvl = blockIdx.y;
  const int tid = threadIdx.x;

  // ---- Stage this level's 128 KB feature table into LDS via async DMA ----
  const float4* gsrc = (const float4*)(table + (size_t)lvl * (TBL * NFEAT));
  float4* ldst = (float4*)lds_tab;
  const int n128 = (TBL * NFEAT) / 4;   // 8192 x b128, 32 per thread
#if __has_builtin(__builtin_amdgcn_global_load_async_to_lds_b128)
  for (int i = tid; i < n128; i += BLOCK) {
    __builtin_amdgcn_global_load_async_to_lds_b128(
        (as1_v4i_ptr)(gsrc + i),
        (as3_v4i_ptr)(ldst + i),
        /*imm offset*/0, /*cpol*/0);
  }
#if __has_builtin(__builtin_amdgcn_s_wait_asynccnt)
  __builtin_amdgcn_s_wait_asynccnt(0);
#else
  asm volatile("s_wait_asynccnt 0" ::: "memory");
#endif
#else
  for (int i = tid; i < n128; i += BLOCK) ldst[i] = gsrc[i];
#endif
  __syncthreads();

  const float NF = NS_F[lvl];           // 1/width (exact grid resolution)
  const float W  = W_F[lvl];            // f32 width, matches jnp 1.0/NS
  const int tileStart = blockIdx.x * PPB;
  const int tileEnd   = min(tileStart + PPB, npts);
  float2* __restrict__ outv = (float2*)out;   // 16 float2 per point row

  for (int p = tileStart + tid; p < tileEnd; p += BLOCK) {
    // speculative prefetch of next batch's coordinates -> global_prefetch_b8
    __builtin_prefetch(inputs + (size_t)(p + BLOCK) * 3, 0, 0);

    const float xr = inputs[3 * (size_t)p + 0];
    const float yr = inputs[3 * (size_t)p + 1];
    const float zr = inputs[3 * (size_t)p + 2];
    const float xx = (xr + 3.0f) * (1.0f / 6.0f);
    const float yy = (yr + 3.0f) * (1.0f / 6.0f);
    const float zz = (zr + 3.0f) * (1.0f / 6.0f);

    const float vx = xx * NF, vy = yy * NF, vz = zz * NF;
    const float mx = floorf(vx), my = floorf(vy), mz = floorf(vz);
    // per = (x - mn*w)/w, with /w folded to *NF (ulp-equivalent, continuous)
    const float px = (xx - mx * W) * NF;
    const float py = (yy - my * W) * NF;
    const float pz = (zz - mz * W) * NF;

    const uint32_t ux = (uint32_t)mx, uy = (uint32_t)my, uz = (uint32_t)mz;
    const uint32_t hx0 = ux;                   const uint32_t hx1 = hx0 + 1u;
    const uint32_t hy0 = uy * 2654435761u;     const uint32_t hy1 = hy0 + 2654435761u;
    const uint32_t hz0 = uz * 805459861u;      const uint32_t hz1 = hz0 + 805459861u;

    const float wx1 = px, wx0 = 1.0f - px;     // corner bit2 -> x
    const float wy1 = py, wy0 = 1.0f - py;     // corner bit1 -> y
    const float wz1 = pz, wz0 = 1.0f - pz;     // corner bit0 -> z

    float acc0 = 0.0f, acc1 = 0.0f;
#define HG_CORNER(bx, by, bz) {                                        \
    const uint32_t idx = (hx##bx ^ hy##by ^ hz##bz) & (TBL - 1u);      \
    const float2 f = lds_tab[idx];                                     \
    const float wgt = wx##bx * wy##by * wz##bz;                        \
    acc0 = fmaf(wgt, f.x, acc0);                                       \
    acc1 = fmaf(wgt, f.y, acc1); }
    HG_CORNER(0,0,0) HG_CORNER(0,0,1) HG_CORNER(0,1,0) HG_CORNER(0,1,1)
    HG_CORNER(1,0,0) HG_CORNER(1,0,1) HG_CORNER(1,1,0) HG_CORNER(1,1,1)
#undef HG_CORNER

    outv[(size_t)p * NLEVELS + lvl] = make_float2(acc0, acc1);
  }
}

extern "C" void kernel_launch(void* const* d_in, const int* in_sizes, int n_in,
                              void* d_out, int out_size, void* d_ws, size_t ws_size,
                              hipStream_t stream) {
  (void)n_in; (void)out_size; (void)d_ws; (void)ws_size;
  const float* inputs = (const float*)d_in[0];   // (N, 3) f32
  const float* table  = (const float*)d_in[1];   // (16, 16384, 2) f32
  float* out = (float*)d_out;                    // (N, 32) f32

  const int npts = in_sizes[0] / 3;
  const size_t shmem = (size_t)TBL * sizeof(float2);   // 128 KB dynamic LDS

  // Opt in to >64 KB dynamic LDS (idempotent; ignore failure on toolchains
  // where the default already allows it).
  (void)hipFuncSetAttribute((const void*)hashgrid_encode_kernel,
                            hipFuncAttributeMaxDynamicSharedMemorySize,
                            (int)shmem);

  const int tiles = (npts + PPB - 1) / PPB;
  dim3 grid((unsigned)tiles, NLEVELS);
  hashgrid_encode_kernel<<<grid, BLOCK, shmem, stream>>>(inputs, table, out, npts);
}